// MaxUnpooling2D_53704271069720
// MI455X (gfx1250) — compile-verified
//
#include <hip/hip_runtime.h>

// MaxUnpooling2D (pool 2x2) scatter-add for MI455X / gfx1250.
//
// inputs : d_in[0] = updates, float32 [16,128,128,64]
//          d_in[1] = mask,    int32   [16,128,128,64] (flat idx into OH*OW*C)
// output : d_out   = float32 [16,256,256,64], zero-initialized then
//          out[b, y, x, f] += updates  with duplicate-index summation.
//
// Key identity (C=64, OW=256 are powers of two):
//   out_offset_in_batch = (mask & ~63) + (flat_elem_idx & 63)
//
// Roofline: 128 MB streaming reads + 256 MB zero-fill + 16.7M L2-resident
// atomics  ->  ~384 MB HBM traffic, ~17 us at 23.3 TB/s. Both kernels are
// pure-bandwidth; this version fattens blocks to amortize prologue overhead.

namespace {
constexpr int kB   = 16;
constexpr int kH   = 128;
constexpr int kW   = 128;
constexpr int kC   = 64;
constexpr int kOH  = kH * 2;
constexpr int kOW  = kW * 2;
constexpr int kEPB = kH * kW * kC;    // 1,048,576 input elements per batch
constexpr int kOPB = kOH * kOW * kC;  // 4,194,304 output elements per batch
}  // namespace

// ---------------------------------------------------------------------------
// Pass 1: zero the output. 4x b128 stores per thread (64 B/lane) so only
// 16384 blocks carry prologue cost. Regular-temporal stores on purpose: the
// dirty zero lines stay resident in L2 (192 MB) where the scatter atomics
// will subsequently hit.
// ---------------------------------------------------------------------------
__global__ __launch_bounds__(256) void zero_f4(float4* __restrict__ o) {
  const float4 z{0.f, 0.f, 0.f, 0.f};
  int i = (blockIdx.x * 256 + threadIdx.x) * 4;
  o[i + 0] = z;
  o[i + 1] = z;
  o[i + 2] = z;
  o[i + 3] = z;
}

// ---------------------------------------------------------------------------
// Native no-return FP32 global atomic add (STOREcnt-tracked, fire-and-forget;
// s_endpgm's implicit wait-idle drains it). Inline asm guarantees the native
// gfx1250 instruction at device scope instead of a CAS loop.
// ---------------------------------------------------------------------------
__device__ __forceinline__ void gatom_add_f32(float* p, float v) {
  asm volatile("global_atomic_add_f32 %0, %1, off scope:SCOPE_DEV"
               :
               : "v"(p), "v"(v)
               : "memory");
}

// ---------------------------------------------------------------------------
// Pass 2: scatter. gridDim.z = batch index so concurrently-resident blocks
// scatter into the same 16.8 MB output slice -> atomics stay in L2.
// Each thread: 2x int4 + 2x float4 coalesced B128 loads, 8 atomics.
// ---------------------------------------------------------------------------
__global__ __launch_bounds__(256) void unpool_scatter(
    const float4* __restrict__ upd, const int4* __restrict__ msk,
    float* __restrict__ out) {
  const int b  = blockIdx.z;
  const int t  = blockIdx.x * 256 + threadIdx.x;  // 0 .. kEPB/8 - 1
  const int i8 = b * (kEPB / 4) + t * 2;          // first vec4 index

  // gfx1250 global_prefetch_b8 ~8KB ahead on both input streams.
  __builtin_prefetch(&upd[i8 + 512], 0, 0);
  __builtin_prefetch(&msk[i8 + 512], 0, 0);

  const float4 u0 = upd[i8 + 0];
  const float4 u1 = upd[i8 + 1];
  const int4   m0 = msk[i8 + 0];
  const int4   m1 = msk[i8 + 1];

  // channel of first element handled by this thread (<= 56, so +k is exact)
  const int f0 = (t << 3) & (kC - 1);

  float* ob = out + (size_t)b * kOPB;
  gatom_add_f32(ob + ((m0.x & ~63) + f0 + 0), u0.x);
  gatom_add_f32(ob + ((m0.y & ~63) + f0 + 1), u0.y);
  gatom_add_f32(ob + ((m0.z & ~63) + f0 + 2), u0.z);
  gatom_add_f32(ob + ((m0.w & ~63) + f0 + 3), u0.w);
  gatom_add_f32(ob + ((m1.x & ~63) + f0 + 4), u1.x);
  gatom_add_f32(ob + ((m1.y & ~63) + f0 + 5), u1.y);
  gatom_add_f32(ob + ((m1.z & ~63) + f0 + 6), u1.z);
  gatom_add_f32(ob + ((m1.w & ~63) + f0 + 7), u1.w);
}

// ---------------------------------------------------------------------------
extern "C" void kernel_launch(void* const* d_in, const int* in_sizes, int n_in,
                              void* d_out, int out_size, void* d_ws,
                              size_t ws_size, hipStream_t stream) {
  const float4* upd = (const float4*)d_in[0];
  const int4*   msk = (const int4*)d_in[1];
  float*        out = (float*)d_out;

  // Zero-fill: out_size = 67,108,864 floats = 16,777,216 float4;
  // 4 float4 per thread, 1024 per block -> 16384 blocks (exact division).
  const int n_zero_blocks = out_size / 4 / (256 * 4);
  zero_f4<<<dim3(n_zero_blocks), dim3(256), 0, stream>>>((float4*)d_out);

  // Scatter: (kEPB/8)/256 = 512 blocks per batch, z = batch (exact division).
  dim3 grid(kEPB / 8 / 256, 1, kB);
  unpool_scatter<<<grid, dim3(256), 0, stream>>>(upd, msk, out);
}